// GNN_88399016887080
// MI455X (gfx1250) — compile-verified
//
#include <hip/hip_runtime.h>

#define N_NODES   100000
#define N_EDGES   1600000
#define FEAT      21
#define KPAD      24
#define EMB       128
#define N_GRAPHS  512
#define BN_EPS    1e-5f

typedef float v8f __attribute__((ext_vector_type(8)));
typedef float v2f __attribute__((ext_vector_type(2)));

static __device__ __forceinline__ void atomicMaxF(float* addr, float val) {
  // Bit-trick float max: int ordering matches float ordering for >=0,
  // unsigned ordering is reversed for negatives.
  if (val >= 0.0f) {
    atomicMax((int*)addr, __float_as_int(val));
  } else {
    atomicMin((unsigned int*)addr, __float_as_uint(val));
  }
}

// ---------------- utility fills ----------------
__global__ void fill_f32(float* __restrict__ p, float v, int n) {
  int i = blockIdx.x * blockDim.x + threadIdx.x;
  if (i < n) p[i] = v;
}

__global__ void copy_f32(const float* __restrict__ s, float* __restrict__ d, int n) {
  int i = blockIdx.x * blockDim.x + threadIdx.x;
  if (i < n) d[i] = s[i];
}

// ---------------- layer-1 input prep ----------------
// hin24[node][c] = x[node][c] for c<21, else 0  (K padded to 24 for WMMA)
__global__ void init_hin24(const float* __restrict__ x, float* __restrict__ h) {
  int i = blockIdx.x * blockDim.x + threadIdx.x;
  if (i >= N_NODES * KPAD) return;
  int node = i / KPAD, c = i - node * KPAD;
  h[i] = (c < FEAT) ? x[node * FEAT + c] : 0.0f;
}

// hin24[dst] += x[src] over 21 features (segment_sum of gathered rows)
__global__ void scatter_feat21(const float* __restrict__ x,
                               const int* __restrict__ src,
                               const int* __restrict__ dst,
                               float* __restrict__ h) {
  int i = blockIdx.x * blockDim.x + threadIdx.x;
  if (i >= N_EDGES * FEAT) return;
  int e = i / FEAT, f = i - e * FEAT;
  atomicAdd(&h[dst[e] * KPAD + f], x[src[e] * FEAT + f]);
}

// W1a (21x128) -> zero-padded (24x128)
__global__ void pad_w1a(const float* __restrict__ w, float* __restrict__ wp) {
  int i = blockIdx.x * blockDim.x + threadIdx.x;
  if (i >= KPAD * EMB) return;
  int r = i / EMB, c = i - r * EMB;
  wp[i] = (r < FEAT) ? w[r * EMB + c] : 0.0f;
}

// ---------------- layer-2 aggregation (128 feats, float4 gather) ----------------
__global__ void scatter_feat128(const float* __restrict__ h,
                                const int* __restrict__ src,
                                const int* __restrict__ dst,
                                float* __restrict__ outh) {
  int i = blockIdx.x * blockDim.x + threadIdx.x;   // N_EDGES * 32 threads
  if (i >= N_EDGES * 32) return;
  // one wave handles one edge; force the uniform edge id into an SGPR so
  // src[e]/dst[e] become scalar (SMEM) loads instead of per-lane VMEM.
  const int e = __builtin_amdgcn_readfirstlane(i >> 5);
  const int q = (i & 31) << 2;
  const float4 v = *(const float4*)(h + src[e] * EMB + q);
  float* base = outh + dst[e] * EMB + q;
  atomicAdd(base + 0, v.x);
  atomicAdd(base + 1, v.y);
  atomicAdd(base + 2, v.z);
  atomicAdd(base + 3, v.w);
}

// ---------------- WMMA GEMM: C[M x 128] = A[M x K] * B[K x 128] + bias ----------------
// One wave -> 16 rows x 128 cols via 8 accumulator tiles of V_WMMA_F32_16X16X4_F32.
// B is staged once per workgroup in LDS, pair-interleaved and swizzled:
//   row-pair p=k/2 at stride 288 floats, column n at 2n+(k&1)
// so a B fragment is ONE conflict-free ds_load_b64 (288 mod 64 == 32 puts the
// upper half-wave's row-pair on the opposite 32 banks). All 8 fragments for a
// k-step are loaded into distinct registers first so the scheduler can issue
// the DS loads as one clause and fire the 8 WMMAs back-to-back.
template <int K, bool RELU>
__global__ __launch_bounds__(256) void gemm_wmma(const float* __restrict__ A,
                                                 const float* __restrict__ B,
                                                 const float* __restrict__ bias,
                                                 float* __restrict__ C, int M) {
  constexpr int PSTRIDE = 288;
  __shared__ float lb[(K / 2) * PSTRIDE];

  const int tid = threadIdx.x;
  for (int idx = tid; idx < K * EMB; idx += 256) {
    const int k = idx >> 7, n = idx & 127;
    lb[(k >> 1) * PSTRIDE + (n << 1) + (k & 1)] = B[idx];
  }
  __syncthreads();

  const int lane = tid & 31;
  const int wave = tid >> 5;
  const int m0 = (blockIdx.x * 8 + wave) * 16;
  if (m0 >= M) return;                     // wave-uniform: EXEC all-ones at WMMA
  const int halfsel = lane >> 4;           // 0: lanes 0-15, 1: lanes 16-31
  const int l16 = lane & 15;

  v8f acc[8];
#pragma unroll
  for (int nt = 0; nt < 8; ++nt) {
    float bv = bias[nt * 16 + l16];        // N depends only on lane
#pragma unroll
    for (int r = 0; r < 8; ++r) acc[nt][r] = bv;
  }

  const float* arow = A + (m0 + l16) * K;  // A fragment: row = lane&15
  __builtin_prefetch(arow, 0, 0);          // global_prefetch_b8 on the A stream
#pragma unroll 2
  for (int k = 0; k < K; k += 4) {
    const int ka = k + halfsel * 2;        // K split across half-waves
    v2f a;
    a.x = arow[ka];
    a.y = arow[ka + 1];
    const float* lrow = &lb[(ka >> 1) * PSTRIDE + (l16 << 1)];
    v2f bfrag[8];
#pragma unroll
    for (int nt = 0; nt < 8; ++nt)         // 4x ds_load_2addr_b64, one wait
      bfrag[nt] = *(const v2f*)&lrow[nt << 5];
#pragma unroll
    for (int nt = 0; nt < 8; ++nt)         // 8 back-to-back WMMAs
      acc[nt] = __builtin_amdgcn_wmma_f32_16x16x4_f32(
          false, a, false, bfrag[nt], (short)0, acc[nt], false, false);
  }

#pragma unroll
  for (int nt = 0; nt < 8; ++nt) {
    const int n = nt * 16 + l16;
#pragma unroll
    for (int r = 0; r < 8; ++r) {
      const int row = m0 + r + halfsel * 8;    // C layout: VGPR r -> M=r / M=8+r
      float v = acc[nt][r];
      if (RELU) v = fmaxf(v, 0.0f);
      C[row * EMB + n] = v;
    }
  }
}

// ---------------- BatchNorm: column sums / sumsq, then apply ----------------
__global__ void bn_stats(const float* __restrict__ Y, float* __restrict__ stats, int M) {
  const int col = threadIdx.x;             // blockDim = 128, col-per-thread: coalesced rows
  const int r0 = blockIdx.x * 256;
  const int r1 = (r0 + 256 < M) ? (r0 + 256) : M;
  float s = 0.0f, ss = 0.0f;
  for (int r = r0; r < r1; ++r) {
    float v = Y[r * EMB + col];
    s += v;
    ss += v * v;
  }
  atomicAdd(&stats[col], s);
  atomicAdd(&stats[EMB + col], ss);
}

__global__ void bn_apply(float* __restrict__ Y, const float* __restrict__ stats,
                         const float* __restrict__ gamma, const float* __restrict__ beta,
                         int M, int relu_flag) {
  int i = blockIdx.x * blockDim.x + threadIdx.x;
  if (i >= M * EMB) return;
  const int col = i & (EMB - 1);
  const float invN = 1.0f / (float)M;
  const float m = stats[col] * invN;
  const float var = stats[EMB + col] * invN - m * m;   // biased, matches jnp.var
  float val = (Y[i] - m) * rsqrtf(var + BN_EPS) * gamma[col] + beta[col];
  if (relu_flag) val = fmaxf(val, 0.0f);
  Y[i] = val;
}

// ---------------- graph readout ----------------
__global__ void readout_scatter(const float* __restrict__ h2, const int* __restrict__ batch,
                                float* __restrict__ gmax, float* __restrict__ gsum,
                                float* __restrict__ cnt) {
  int i = blockIdx.x * blockDim.x + threadIdx.x;     // N_NODES * EMB
  if (i >= N_NODES * EMB) return;
  const int node = i >> 7, col = i & 127;
  const int g = batch[node];
  const float v = h2[i];
  atomicMaxF(&gmax[g * EMB + col], v);
  atomicAdd(&gsum[g * EMB + col], v);
  if (col == 0) atomicAdd(&cnt[g], 1.0f);
}

__global__ void readout_finalize(const float* __restrict__ gmax, const float* __restrict__ gsum,
                                 const float* __restrict__ cnt, float* __restrict__ out) {
  int i = blockIdx.x * blockDim.x + threadIdx.x;     // N_GRAPHS * 2*EMB
  if (i >= N_GRAPHS * 2 * EMB) return;
  const int g = i / (2 * EMB), c = i - g * (2 * EMB);
  out[i] = (c < EMB) ? gmax[g * EMB + c]
                     : gsum[g * EMB + (c - EMB)] / fmaxf(cnt[g], 1.0f);
}

// ---------------- host orchestration ----------------
static inline int cdiv(int a, int b) { return (a + b - 1) / b; }

extern "C" void kernel_launch(void* const* d_in, const int* in_sizes, int n_in,
                              void* d_out, int out_size, void* d_ws, size_t ws_size,
                              hipStream_t stream) {
  const float* x     = (const float*)d_in[0];
  const int*   eidx  = (const int*)d_in[1];
  const int*   src   = eidx;                 // edge_index[0]
  const int*   dst   = eidx + N_EDGES;       // edge_index[1]
  const int*   batch = (const int*)d_in[2];
  const float* W1a = (const float*)d_in[4];
  const float* b1a = (const float*)d_in[5];
  const float* g1a = (const float*)d_in[6];
  const float* bt1a = (const float*)d_in[7];
  const float* W1b = (const float*)d_in[8];
  const float* b1b = (const float*)d_in[9];
  const float* W2a = (const float*)d_in[10];
  const float* b2a = (const float*)d_in[11];
  const float* g2a = (const float*)d_in[12];
  const float* bt2a = (const float*)d_in[13];
  const float* W2b = (const float*)d_in[14];
  const float* b2b = (const float*)d_in[15];
  const float* g_bn1 = (const float*)d_in[16];
  const float* b_bn1 = (const float*)d_in[17];
  const float* g_bn2 = (const float*)d_in[18];
  const float* b_bn2 = (const float*)d_in[19];
  float* out = (float*)d_out;

  // workspace carve-up (floats)
  float* ws    = (float*)d_ws;
  float* hin24 = ws;                         // 100000*24   = 2,400,000
  float* wpad  = hin24 + N_NODES * KPAD;     // 24*128      = 3,072
  float* bufA  = wpad + KPAD * EMB;          // 100000*128  = 12,800,000
  float* bufB  = bufA + N_NODES * EMB;       // 100000*128  = 12,800,000
  float* stats = bufB + N_NODES * EMB;       // 256
  float* gmax  = stats + 2 * EMB;            // 512*128     = 65,536
  float* gsum  = gmax + N_GRAPHS * EMB;      // 512*128     = 65,536
  float* cnt   = gsum + N_GRAPHS * EMB;      // 512

  const int BT = 256;
  const int NE = N_NODES * EMB;              // 12,800,000
  const int gemm_blocks = cdiv(cdiv(N_NODES, 16), 8);   // 782
  const int stat_blocks = cdiv(N_NODES, 256);           // 391

  // ===== Layer 1 =====
  init_hin24<<<cdiv(N_NODES * KPAD, BT), BT, 0, stream>>>(x, hin24);
  scatter_feat21<<<cdiv(N_EDGES * FEAT, BT), BT, 0, stream>>>(x, src, dst, hin24);
  pad_w1a<<<cdiv(KPAD * EMB, BT), BT, 0, stream>>>(W1a, wpad);

  gemm_wmma<KPAD, false><<<gemm_blocks, BT, 0, stream>>>(hin24, wpad, b1a, bufA, N_NODES);
  fill_f32<<<1, 256, 0, stream>>>(stats, 0.0f, 2 * EMB);
  bn_stats<<<stat_blocks, EMB, 0, stream>>>(bufA, stats, N_NODES);
  bn_apply<<<cdiv(NE, BT), BT, 0, stream>>>(bufA, stats, g1a, bt1a, N_NODES, 1);

  gemm_wmma<EMB, true><<<gemm_blocks, BT, 0, stream>>>(bufA, W1b, b1b, bufB, N_NODES);
  fill_f32<<<1, 256, 0, stream>>>(stats, 0.0f, 2 * EMB);
  bn_stats<<<stat_blocks, EMB, 0, stream>>>(bufB, stats, N_NODES);
  bn_apply<<<cdiv(NE, BT), BT, 0, stream>>>(bufB, stats, g_bn1, b_bn1, N_NODES, 1);
  // bufB = h (layer-1 output)

  // ===== Layer 2 =====
  copy_f32<<<cdiv(NE, BT), BT, 0, stream>>>(bufB, bufA, NE);                 // hin2 = h
  scatter_feat128<<<cdiv(N_EDGES * 32, BT), BT, 0, stream>>>(bufB, src, dst, bufA); // += agg

  gemm_wmma<EMB, false><<<gemm_blocks, BT, 0, stream>>>(bufA, W2a, b2a, bufB, N_NODES);
  fill_f32<<<1, 256, 0, stream>>>(stats, 0.0f, 2 * EMB);
  bn_stats<<<stat_blocks, EMB, 0, stream>>>(bufB, stats, N_NODES);
  bn_apply<<<cdiv(NE, BT), BT, 0, stream>>>(bufB, stats, g2a, bt2a, N_NODES, 1);

  gemm_wmma<EMB, true><<<gemm_blocks, BT, 0, stream>>>(bufB, W2b, b2b, bufA, N_NODES);
  fill_f32<<<1, 256, 0, stream>>>(stats, 0.0f, 2 * EMB);
  bn_stats<<<stat_blocks, EMB, 0, stream>>>(bufA, stats, N_NODES);
  bn_apply<<<cdiv(NE, BT), BT, 0, stream>>>(bufA, stats, g_bn2, b_bn2, N_NODES, 0);
  // bufA = h2

  // ===== Readout =====
  fill_f32<<<cdiv(N_GRAPHS * EMB, BT), BT, 0, stream>>>(gmax, -INFINITY, N_GRAPHS * EMB);
  fill_f32<<<cdiv(N_GRAPHS * EMB + N_GRAPHS, BT), BT, 0, stream>>>(gsum, 0.0f,
                                                                   N_GRAPHS * EMB + N_GRAPHS);
  readout_scatter<<<cdiv(NE, BT), BT, 0, stream>>>(bufA, batch, gmax, gsum, cnt);
  readout_finalize<<<cdiv(N_GRAPHS * 2 * EMB, BT), BT, 0, stream>>>(gmax, gsum, cnt, out);
}